// AdvancedCTCDecoder_46703474377102
// MI455X (gfx1250) — compile-verified
//
#include <hip/hip_runtime.h>
#include <hip/hip_bf16.h>
#include <stdint.h>

// CTC greedy decode for MI455X (gfx1250).
// log_probs [32,4096,1025] fp32 -> preds/keep/max_logp, each [32,4096].
// Pure HBM-bandwidth problem (~537 MB streamed, ~23 us floor at 23.3 TB/s).
// CDNA5 path: async global->LDS engine (ASYNCcnt) stages each row in LDS;
// staged counted waits (async-done is in-order per ISA 08 §4.1) pipeline the
// argmax scan against in-flight HBM transfers; wave32 shuffle reduction.

#define BATCH 32
#define T_LEN 4096
#define VOCAB 1025
#define BLANK 1024
#define BT (BATCH * T_LEN)

// ---------------------------------------------------------------------------
// Kernel 1: per-row argmax. One wave32 per (b,t) row. 8 waves per block.
// ---------------------------------------------------------------------------
__global__ __launch_bounds__(256) void ctc_argmax_kernel(
    const float* __restrict__ lp,
    float* __restrict__ out_pred,   // [BT] argmax index, stored as float
    float* __restrict__ out_logp)   // [BT] max log-prob
{
    // 8 waves * 1024 floats = 32 KB LDS per block (320 KB/WGP available).
    __shared__ float smem[8 * 1024];

    const int lane = threadIdx.x & 31;
    const int wave = threadIdx.x >> 5;
    const int row  = blockIdx.x * 8 + wave;          // grid sized exactly: row < BT

    const float* rowp = lp + (size_t)row * VOCAB;
    float* wbuf = &smem[wave * 1024];

    // LDS byte offset: flat shared-aperture address truncated to 32 bits is the
    // LDS offset (ISA 10.2).
    const unsigned lds_base = (unsigned)(uintptr_t)wbuf;

    // Issue 8 async b128 loads up front: iteration i moves floats
    // [i*128 + lane*4 .. +3] (each instruction = 512 contiguous bytes/wave).
    // Tracked on ASYNCcnt; lands directly in LDS, no VGPR staging.
#pragma unroll
    for (int i = 0; i < 8; ++i) {
        const unsigned           loff  = lds_base + (unsigned)((i * 128 + lane * 4) * 4);
        const unsigned long long gaddr =
            (unsigned long long)(uintptr_t)(rowp + i * 128 + lane * 4);
        asm volatile("global_load_async_to_lds_b128 %0, %1, off"
                     :: "v"(loff), "v"(gaddr)
                     : "memory");
    }

    // Element 1024 (the odd one out) via a normal global load, overlapped
    // with the async transfers (tracked on LOADcnt, independent).
    const float last = rowp[1024];

    // Each lane scans its 32 elements (ascending index, strict '>' so the
    // FIRST occurrence of the max wins, matching jnp.argmax).
    // Async loads complete in order, so chunk i is resident in LDS once
    // ASYNCcnt <= 7-i: consume chunk i as soon as it lands (software pipeline).
    float m   = -__builtin_inff();
    int   idx = 0;
#pragma unroll
    for (int i = 0; i < 8; ++i) {
        asm volatile("s_wait_asynccnt %0" :: "i"(7 - i) : "memory");
        const int    base = i * 128 + lane * 4;
        const float4 v    = *reinterpret_cast<const float4*>(&wbuf[base]); // ds_load_b128
        if (v.x > m) { m = v.x; idx = base + 0; }
        if (v.y > m) { m = v.y; idx = base + 1; }
        if (v.z > m) { m = v.z; idx = base + 2; }
        if (v.w > m) { m = v.w; idx = base + 3; }
    }

    // Wave32 butterfly reduction: larger value wins; on ties, smaller index.
#pragma unroll
    for (int d = 16; d >= 1; d >>= 1) {
        const float om = __shfl_xor(m,   d, 32);
        const int   oi = __shfl_xor(idx, d, 32);
        if (om > m || (om == m && oi < idx)) { m = om; idx = oi; }
    }

    if (lane == 0) {
        // Element 1024 has the largest index, so strict '>' preserves
        // first-occurrence tie-breaking.
        if (last > m) { m = last; idx = 1024; }
        out_pred[row] = (float)idx;   // exact: idx <= 1024 << 2^24
        out_logp[row] = m;
    }
}

// ---------------------------------------------------------------------------
// Kernel 2: CTC collapse mask. keep = (pred!=BLANK) && (pred!=prev) && (t<len).
// Reads the float preds written by kernel 1 (exact integers in fp32).
// ---------------------------------------------------------------------------
__global__ __launch_bounds__(256) void ctc_keep_kernel(
    const float* __restrict__ pred_f,   // [BT]
    const int*  __restrict__ lengths,   // [BATCH]
    float* __restrict__ out_keep)       // [BT]
{
    const int i = blockIdx.x * blockDim.x + threadIdx.x;
    if (i >= BT) return;
    const int b = i / T_LEN;
    const int t = i - b * T_LEN;

    const int pred = (int)pred_f[i];
    const int prev = (t > 0) ? (int)pred_f[i - 1] : -1;   // -1 sentinel at t=0
    const int len  = lengths[b];

    const bool keep = (pred != BLANK) && (pred != prev) && (t < len);
    out_keep[i] = keep ? 1.0f : 0.0f;
}

// ---------------------------------------------------------------------------
// Harness entry point.
// ---------------------------------------------------------------------------
extern "C" void kernel_launch(void* const* d_in, const int* in_sizes, int n_in,
                              void* d_out, int out_size, void* d_ws, size_t ws_size,
                              hipStream_t stream) {
    (void)in_sizes; (void)n_in; (void)out_size; (void)d_ws; (void)ws_size;

    const float* lp   = (const float*)d_in[0];   // log_probs [32,4096,1025] fp32
    const int*   lens = (const int*)d_in[1];     // input_lengths [32]

    float* out      = (float*)d_out;
    float* out_pred = out;              // [BT] preds (as float)
    float* out_keep = out + BT;         // [BT] keep mask (0.0 / 1.0)
    float* out_logp = out + 2 * BT;     // [BT] max log-prob

    // Kernel 1: 131072 rows, 8 waves (rows) per 256-thread block.
    ctc_argmax_kernel<<<BT / 8, 256, 0, stream>>>(lp, out_pred, out_logp);

    // Kernel 2: elementwise collapse mask (depends on out_pred).
    ctc_keep_kernel<<<(BT + 255) / 256, 256, 0, stream>>>(out_pred, lens, out_keep);
}